// CircuitEncoder_71665824301416
// MI455X (gfx1250) — compile-verified
//
#include <hip/hip_runtime.h>
#include <hip/hip_bf16.h>

typedef __attribute__((ext_vector_type(2))) float v2f;
typedef __attribute__((ext_vector_type(8))) float v8f;

#define NPN    1024
#define DFEAT  128
#define NB     512
#define NE     2048
#define NNODES (NB * NPN)            // 524288
#define NEDGES (NB * NE)             // 1048576
#define ALLMASK 0x7FFFFFFF
#define LDSP   132                   // padded LDS row stride (floats): banks = (4r+k)%64

// ---------------- degree / normalization ----------------

__global__ void k_deg_init(float* __restrict__ deg) {
    int i = blockIdx.x * blockDim.x + threadIdx.x;
    if (i < NNODES) deg[i] = 1.0f;   // self-loop contributes 1
}

__global__ void k_deg_accum(const int* __restrict__ ei, float* __restrict__ deg) {
    int ge = blockIdx.x * blockDim.x + threadIdx.x;
    if (ge >= NEDGES) return;
    int b = ge >> 11;                // / NE
    int e = ge & (NE - 1);
    int c = ei[b * 2 * NE + NE + e] + b * NPN;
    unsafeAtomicAdd(&deg[c], 1.0f);
}

__global__ void k_deg_finalize(float* __restrict__ deg) {
    int i = blockIdx.x * blockDim.x + threadIdx.x;
    if (i < NNODES) deg[i] = rsqrtf(deg[i]);   // deg >= 1 always
}

// ---------------- WMMA fp32 GEMM: H[M x 128] = X[M x 128] @ W[128 x 128] ----
// One block = 8 waves covering 16 rows x 128 cols; wave w owns N-tile w.
// A tile (16x128) staged in LDS row-major; W staged TRANSPOSED (Wt[n][k]) so
// both fragments are contiguous k-pairs -> single ds_load_b64 each.
// A frag (16x4 f32, 2 VGPR): lane L holds A[L&15][2*(L>>4)+{0,1}]
// B frag (4x16 f32, 2 VGPR): lane L holds B[2*(L>>4)+{0,1}][L&15]
// D (16x16 f32, 8 VGPR):     elem v of lane L = D[v + 8*(L>>4)][L&15]

__global__ void __launch_bounds__(256) k_gemm_wmma(const float* __restrict__ X,
                                                   const float* __restrict__ W,
                                                   float* __restrict__ H) {
    __shared__ float Wt[DFEAT * LDSP];   // W transposed, padded: ~66 KB
    __shared__ float As[16 * LDSP];      // A tile, padded: ~8.25 KB

    // stage W transposed: Wt[n*LDSP + k] = W[k*DFEAT + n]
    for (int idx = threadIdx.x; idx < DFEAT * DFEAT / 4; idx += 256) {
        int k = idx >> 5;                 // 0..127
        int n4 = (idx & 31) * 4;          // 0..124 step 4
        float4 w = *(const float4*)(W + k * DFEAT + n4);
        Wt[(n4 + 0) * LDSP + k] = w.x;
        Wt[(n4 + 1) * LDSP + k] = w.y;
        Wt[(n4 + 2) * LDSP + k] = w.z;
        Wt[(n4 + 3) * LDSP + k] = w.w;
    }

    // stage A tile: 16 rows x 128 floats = 512 float4s, 2 per thread
    const int mBase = blockIdx.x * 16;
    for (int idx = threadIdx.x; idx < 512; idx += 256) {
        int r  = idx >> 5;                // 0..15
        int k4 = (idx & 31) * 4;          // 0..124 step 4
        float4 a = *(const float4*)(X + (size_t)(mBase + r) * DFEAT + k4);
        *(float4*)(As + r * LDSP + k4) = a;   // LDSP%4==0 -> 16B aligned
    }
    __syncthreads();

    const int wave  = threadIdx.x >> 5;
    const int lane  = threadIdx.x & 31;
    const int nBase = wave * 16;
    const int col   = lane & 15;
    const int kHalf = (lane >> 4) * 2;    // 0 or 2

    const float* aRow = As + col * LDSP + kHalf;            // A[row=col][k..]
    const float* bCol = Wt + (nBase + col) * LDSP + kHalf;  // B[k..][n=col]

    v8f acc = {};
    #pragma unroll
    for (int k = 0; k < DFEAT; k += 4) {
        v2f a = *(const v2f*)(aRow + k);   // ds_load_b64, 8B aligned
        v2f b = *(const v2f*)(bCol + k);   // ds_load_b64, 8B aligned
        acc = __builtin_amdgcn_wmma_f32_16x16x4_f32(
                  false, a, false, b, (short)0, acc, false, false);
    }

    #pragma unroll
    for (int v = 0; v < 8; ++v) {
        int r = mBase + v + ((lane >> 4) << 3);
        H[(size_t)r * DFEAT + nBase + col] = acc[v];
    }
}

// ---------------- scatter init: self-loop term + bias ----------------

__global__ void k_scatter_init(const float* __restrict__ H,
                               const float* __restrict__ dinv,
                               const float* __restrict__ bias,
                               float* __restrict__ out, int rowMask) {
    unsigned idx = blockIdx.x * blockDim.x + threadIdx.x;  // < NNODES*DFEAT
    int i = idx >> 7;
    int k = idx & (DFEAT - 1);
    float dv = dinv[i];
    out[idx] = H[(size_t)(i & rowMask) * DFEAT + k] * dv * dv + bias[k];
}

// ---------------- edge scatter: one wave32 per edge ----------------

__global__ void __launch_bounds__(256) k_scatter_edges(const float* __restrict__ H,
                                                       const float* __restrict__ dinv,
                                                       const int* __restrict__ ei,
                                                       float* __restrict__ out,
                                                       int rowMask) {
    int ge   = (blockIdx.x * blockDim.x + threadIdx.x) >> 5;
    int lane = threadIdx.x & 31;
    if (ge >= NEDGES) return;
    int b = ge >> 11;
    int e = ge & (NE - 1);
    int r = ei[b * 2 * NE + e]      + b * NPN;
    int c = ei[b * 2 * NE + NE + e] + b * NPN;
    float norm = dinv[r] * dinv[c];

    const float4* hv = (const float4*)(H + (size_t)(r & rowMask) * DFEAT);
    float4 v = hv[lane];
    float* o = out + (size_t)c * DFEAT + lane * 4;
    unsafeAtomicAdd(o + 0, v.x * norm);
    unsafeAtomicAdd(o + 1, v.y * norm);
    unsafeAtomicAdd(o + 2, v.z * norm);
    unsafeAtomicAdd(o + 3, v.w * norm);
}

// ---------------- ReLU ----------------

__global__ void k_relu(float* __restrict__ x) {
    int i = blockIdx.x * blockDim.x + threadIdx.x;   // over NNODES*DFEAT/4
    float4* p = (float4*)x;
    float4 v = p[i];
    v.x = fmaxf(v.x, 0.0f);
    v.y = fmaxf(v.y, 0.0f);
    v.z = fmaxf(v.z, 0.0f);
    v.w = fmaxf(v.w, 0.0f);
    p[i] = v;
}

// ---------------- launch ----------------

extern "C" void kernel_launch(void* const* d_in, const int* in_sizes, int n_in,
                              void* d_out, int out_size, void* d_ws, size_t ws_size,
                              hipStream_t stream) {
    const int*   edge = (const int*)  d_in[0];
    const float* emb  = (const float*)d_in[1];
    const float* W1   = (const float*)d_in[2];
    const float* b1   = (const float*)d_in[3];
    const float* W2   = (const float*)d_in[4];
    const float* b2   = (const float*)d_in[5];
    const float* W3   = (const float*)d_in[6];
    const float* b3   = (const float*)d_in[7];
    float* dout = (float*)d_out;

    float* bufA = (float*)d_ws;                         // NNODES*DFEAT floats (256 MB)
    float* dinv = bufA + (size_t)NNODES * DFEAT;        // NNODES floats (2 MB)
    float* h1   = dinv + NNODES;                        // NPN*DFEAT floats (512 KB)

    const int T = 256;
    const int gNodes  = (NNODES + T - 1) / T;           // 2048
    const int gEdges  = NEDGES / T;                     // 4096
    const int gElems  = NNODES * DFEAT / T;             // 262144
    const int gEdgeWv = NEDGES * 32 / T;                // 131072
    const int gRelu   = NNODES * DFEAT / 4 / T;         // 65536

    // degrees -> dinv (shared across layers)
    k_deg_init    <<<gNodes, T, 0, stream>>>(dinv);
    k_deg_accum   <<<gEdges, T, 0, stream>>>(edge, dinv);
    k_deg_finalize<<<gNodes, T, 0, stream>>>(dinv);

    // ----- layer 1: x is tiled embeddings -> h1 is only 1024x128 -----
    k_gemm_wmma    <<<NPN / 16, T, 0, stream>>>(emb, W1, h1);
    k_scatter_init <<<gElems, T, 0, stream>>>(h1, dinv, b1, bufA, NPN - 1);
    k_scatter_edges<<<gEdgeWv, T, 0, stream>>>(h1, dinv, edge, bufA, NPN - 1);
    k_relu         <<<gRelu, T, 0, stream>>>(bufA);

    // ----- layer 2: bufA -> h2 in d_out -> out back into bufA -----
    k_gemm_wmma    <<<NNODES / 16, T, 0, stream>>>(bufA, W2, dout);
    k_scatter_init <<<gElems, T, 0, stream>>>(dout, dinv, b2, bufA, ALLMASK);
    k_scatter_edges<<<gEdgeWv, T, 0, stream>>>(dout, dinv, edge, bufA, ALLMASK);
    k_relu         <<<gRelu, T, 0, stream>>>(bufA);

    // ----- layer 3: bufA -> h3 in d_out -> out into bufA -> copy to d_out -----
    k_gemm_wmma    <<<NNODES / 16, T, 0, stream>>>(bufA, W3, dout);
    k_scatter_init <<<gElems, T, 0, stream>>>(dout, dinv, b3, bufA, ALLMASK);
    k_scatter_edges<<<gEdgeWv, T, 0, stream>>>(dout, dinv, edge, bufA, ALLMASK);
    k_relu         <<<gRelu, T, 0, stream>>>(bufA);

    hipMemcpyAsync(d_out, bufA, (size_t)NNODES * DFEAT * sizeof(float),
                   hipMemcpyDeviceToDevice, stream);
}